// CLAPP_SNN_11467562680682
// MI455X (gfx1250) — compile-verified
//
#include <hip/hip_runtime.h>

typedef __attribute__((ext_vector_type(2))) float v2f;
typedef __attribute__((ext_vector_type(8))) float v8f;

#define T_STEPS 32
#define HID 4096
#define NOUT 10
#define BETA 0.75f
#define THRESH 1.0f

// C[32,4096] = A[32,4096] x W^T ; W row-major [N=4096][K=4096].
// Block = 256 threads = 8 waves. Wave (ntile, khalf): ntile in 0..3 picks a
// 16-wide N slice, khalf in 0..1 picks a 2048-wide K half. Each wave computes
// both 16-row M tiles so every W fragment feeds two WMMAs. K halves combined
// deterministically through LDS. Grid = 4096/64 = 64 blocks -> 512 waves.
__global__ __launch_bounds__(256) void snn_gemm_f32(const float* __restrict__ A,
                                                    const float* __restrict__ W,
                                                    float* __restrict__ C) {
    __shared__ float red[2048];  // partials from khalf==1 waves (8 tiles x 256)

    const int lane  = threadIdx.x & 31;
    const int wave  = threadIdx.x >> 5;
    const int ntile = wave & 3;
    const int khalf = wave >> 2;
    const int n_base = blockIdx.x * 64 + ntile * 16;

    const int row16 = lane & 15;
    const int koff  = (lane >> 4) << 1;  // lanes 0-15 -> K+0..1, lanes 16-31 -> K+2..3

    const float* Arow0 = A + row16 * HID;         // M tile 0 (rows 0..15)
    const float* Arow1 = A + (16 + row16) * HID;  // M tile 1 (rows 16..31)
    const float* Wrow  = W + (n_base + row16) * HID;

    v8f c0 = {};
    v8f c1 = {};
    int k0 = khalf * 2048 + koff;
#pragma unroll 4
    for (int it = 0; it < 512; ++it, k0 += 4) {
        v2f a0 = *(const v2f*)(Arow0 + k0);
        v2f a1 = *(const v2f*)(Arow1 + k0);
        v2f b  = *(const v2f*)(Wrow + k0);
        c0 = __builtin_amdgcn_wmma_f32_16x16x4_f32(false, a0, false, b, (short)0, c0,
                                                   false, false);
        c1 = __builtin_amdgcn_wmma_f32_16x16x4_f32(false, a1, false, b, (short)0, c1,
                                                   false, false);
    }

    if (khalf == 1) {
#pragma unroll
        for (int r = 0; r < 8; ++r) {
            red[(ntile * 2 + 0) * 256 + r * 32 + lane] = c0[r];
            red[(ntile * 2 + 1) * 256 + r * 32 + lane] = c1[r];
        }
    }
    __syncthreads();
    if (khalf == 0) {
#pragma unroll
        for (int r = 0; r < 8; ++r) {
            c0[r] += red[(ntile * 2 + 0) * 256 + r * 32 + lane];
            c1[r] += red[(ntile * 2 + 1) * 256 + r * 32 + lane];
        }
        // C/D layout: VGPR r -> M = r (lanes 0-15) / M = r+8 (lanes 16-31), N = lane%16
        const int col  = n_base + (lane & 15);
        const int rofs = (lane >> 4) << 3;  // 0 or 8
#pragma unroll
        for (int r = 0; r < 8; ++r) {
            C[(r + rofs) * HID + col]      = c0[r];
            C[(16 + r + rofs) * HID + col] = c1[r];
        }
    }
}

// LIF membrane scan for a hidden layer: parallel over h, serial over t.
// mem = beta*mem + cur; spk = mem > 1; mem -= spk; record post-reset mem.
__global__ __launch_bounds__(256) void lif_hidden(const float* __restrict__ cur,
                                                  float* __restrict__ spk,
                                                  float* __restrict__ memhis,
                                                  int layer) {
    const int h = blockIdx.x * 256 + threadIdx.x;
    float mem = 0.0f;
#pragma unroll
    for (int t = 0; t < T_STEPS; ++t) {
        mem = BETA * mem + cur[t * HID + h];
        const float s = (mem > THRESH) ? 1.0f : 0.0f;
        mem -= s * THRESH;
        spk[t * HID + h]                  = s;
        memhis[(t * 3 + layer) * HID + h] = mem;
    }
}

// Readout currents: 320 dot products of length 4096, one block each.
__global__ __launch_bounds__(256) void out_gemv(const float* __restrict__ spk,
                                                const float* __restrict__ ow,
                                                float* __restrict__ curo) {
    __shared__ float red[256];
    const int t = blockIdx.x / NOUT;
    const int o = blockIdx.x % NOUT;
    const float* s = spk + t * HID;
    const float* w = ow + o * HID;
    float acc = 0.0f;
#pragma unroll 4
    for (int k = threadIdx.x; k < HID; k += 256) acc += s[k] * w[k];
    red[threadIdx.x] = acc;
    __syncthreads();
    for (int off = 128; off > 0; off >>= 1) {
        if (threadIdx.x < off) red[threadIdx.x] += red[threadIdx.x + off];
        __syncthreads();
    }
    if (threadIdx.x == 0) curo[t * NOUT + o] = red[0];
}

// Output-layer LIF scan; records only spikes.
__global__ void lif_out(const float* __restrict__ curo, float* __restrict__ outspk) {
    const int o = threadIdx.x;
    if (o >= NOUT) return;
    float mem = 0.0f;
    for (int t = 0; t < T_STEPS; ++t) {
        mem = BETA * mem + curo[t * NOUT + o];
        const float s = (mem > THRESH) ? 1.0f : 0.0f;
        outspk[t * NOUT + o] = s;
        mem -= s * THRESH;
    }
}

extern "C" void kernel_launch(void* const* d_in, const int* in_sizes, int n_in,
                              void* d_out, int out_size, void* d_ws, size_t ws_size,
                              hipStream_t stream) {
    const float* inp = (const float*)d_in[0];
    const float* fc0 = (const float*)d_in[1];
    const float* fc1 = (const float*)d_in[2];
    const float* fc2 = (const float*)d_in[3];
    const float* ow  = (const float*)d_in[4];
    // d_in[5]=target, d_in[6]=bf: training-only, unused on the eval path.

    float* out    = (float*)d_out;           // [32,10] spikes then [32,3,4096] mems
    float* memhis = out + T_STEPS * NOUT;

    float* cur  = (float*)d_ws;              // [32,4096] currents   (512 KB)
    float* spk  = cur + T_STEPS * HID;       // [32,4096] spikes     (512 KB)
    float* curo = spk + T_STEPS * HID;       // [32,10]   readout currents

    const dim3 gGemm(HID / 64), bGemm(256);
    const dim3 gLif(HID / 256), bLif(256);

    snn_gemm_f32<<<gGemm, bGemm, 0, stream>>>(inp, fc0, cur);
    lif_hidden<<<gLif, bLif, 0, stream>>>(cur, spk, memhis, 0);
    snn_gemm_f32<<<gGemm, bGemm, 0, stream>>>(spk, fc1, cur);
    lif_hidden<<<gLif, bLif, 0, stream>>>(cur, spk, memhis, 1);
    snn_gemm_f32<<<gGemm, bGemm, 0, stream>>>(spk, fc2, cur);
    lif_hidden<<<gLif, bLif, 0, stream>>>(cur, spk, memhis, 2);
    out_gemv<<<dim3(T_STEPS * NOUT), dim3(256), 0, stream>>>(spk, ow, curo);
    lif_out<<<dim3(1), dim3(32), 0, stream>>>(curo, out);
}